// SphereLoRA_86251533238776
// MI455X (gfx1250) — compile-verified
//
#include <hip/hip_runtime.h>
#include <hip/hip_bf16.h>

typedef __attribute__((ext_vector_type(2))) float v2f;
typedef __attribute__((ext_vector_type(8))) float v8f;

// Problem dims (fixed by reference setup_inputs)
#define BB 4
#define CC 320
#define OO 320
#define HH 64
#define WW 64
#define HWPIX 4096            // 64*64
#define NTOT  16384           // B*H*W
#define KCONV 2880            // C*9

#if defined(__has_builtin)
#if __has_builtin(__builtin_amdgcn_global_load_async_to_lds_b32) && \
    __has_builtin(__builtin_amdgcn_s_wait_asynccnt)
#define HAVE_ASYNC_LDS 1
#endif
#endif

typedef __attribute__((address_space(1))) int as1_int;
typedef __attribute__((address_space(3))) int as3_int;

// ---------------------------------------------------------------------------
// Kernel 1: bilinear gather tables. offset is (9,2,64,64) flattened (18,64,64).
// ---------------------------------------------------------------------------
__global__ void build_tables_kernel(const float* __restrict__ off,
                                    int4* __restrict__ tabI,
                                    float4* __restrict__ tabW) {
    int tid = blockIdx.x * blockDim.x + threadIdx.x;
    if (tid >= 9 * HWPIX) return;
    int k   = tid >> 12;
    int pix = tid & 4095;
    int y = pix >> 6, xq = pix & 63;
    float dy = off[((2 * k    ) << 12) + pix];
    float dx = off[((2 * k + 1) << 12) + pix];
    float py = (float)y  - 1.0f + (float)(k / 3) + dy;
    float px = (float)xq - 1.0f + (float)(k % 3) + dx;
    float y0f = floorf(py), x0f = floorf(px);
    float wy = py - y0f, wx = px - x0f;
    int y0 = (int)y0f, x0 = (int)x0f;

    int   id[4];
    float w[4];
#pragma unroll
    for (int j = 0; j < 4; ++j) {
        int yi = y0 + (j >> 1);
        int xi = x0 + (j & 1);
        float wgt = ((j >> 1) ? wy : (1.0f - wy)) * ((j & 1) ? wx : (1.0f - wx));
        bool valid = ((unsigned)yi < 64u) && ((unsigned)xi < 64u);
        int yc = yi < 0 ? 0 : (yi > 63 ? 63 : yi);
        int xc = xi < 0 ? 0 : (xi > 63 ? 63 : xi);
        id[j] = (yc << 6) + xc;
        w[j]  = valid ? wgt : 0.0f;
    }
    tabI[tid] = make_int4(id[0], id[1], id[2], id[3]);
    tabW[tid] = make_float4(w[0], w[1], w[2], w[3]);
}

// ---------------------------------------------------------------------------
// Kernel 2: reorder weight (O,C,3,3) -> Wrt[k][o] with k = kk*320 + c.
// ---------------------------------------------------------------------------
__global__ void reorder_weight_kernel(const float* __restrict__ w,
                                      float* __restrict__ Wrt) {
    int tid = blockIdx.x * blockDim.x + threadIdx.x;   // kidx*320 + o
    if (tid >= KCONV * OO) return;
    int o    = tid % OO;
    int kidx = tid / OO;
    int kk = kidx / CC;
    int c  = kidx % CC;
    Wrt[tid] = w[(size_t)(o * CC + c) * 9 + kk];
}

// ---------------------------------------------------------------------------
// Kernel 3: Mt[k][o] = (w_up @ w_down)[o][k],  k in [0,640), o in [0,320)
// ---------------------------------------------------------------------------
__global__ void build_Mt_kernel(const float* __restrict__ w_down,
                                const float* __restrict__ w_up,
                                float* __restrict__ Mt) {
    int tid = blockIdx.x * blockDim.x + threadIdx.x;   // k*320 + o
    if (tid >= 640 * OO) return;
    int o = tid % OO;
    int k = tid / OO;
    float s = 0.0f;
    for (int l = 0; l < 320; ++l)
        s += w_up[o * 320 + l] * w_down[l * 640 + k];
    Mt[tid] = s;
}

// ---------------------------------------------------------------------------
// Implicit-GEMM conv via V_WMMA_F32_16X16X4_F32.
// Block: 640 threads = 20 waves; wave w owns M-rows [16w,16w+16).
// N-tile = 32 per wave (two 16-col accumulators share one A fragment).
// K = 2880 in 45 chunks of 64 (one 3x3 tap kk per chunk); B chunk in LDS.
// ---------------------------------------------------------------------------
template <int DEFORM>
__global__ __launch_bounds__(640)
void conv_gemm_kernel(const float* __restrict__ x,
                      const float* __restrict__ Wrt,
                      const float* __restrict__ bias,
                      const int4* __restrict__ tabI,
                      const float4* __restrict__ tabW,
                      float* __restrict__ dst) {
    __shared__ float Blds[32][68];                  // [n][k], pad 68 vs banks
    const int tid  = threadIdx.x;
    const int lane = tid & 31;
    const int wave = tid >> 5;                      // 0..19
    const int half = lane >> 4;                     // ISA lane-half
    const int l15  = lane & 15;
    const int o0   = wave << 4;
    const int n0   = blockIdx.x << 5;               // 512 N-tiles of 32
    const int b0   = n0 >> 12;                      // tiles never cross batch
    const int pix0 = n0 & 4095;

    v8f acc0 = {}, acc1 = {};
    for (int ch = 0; ch < 45; ++ch) {
        const int kk = ch / 5;
        const int c0 = (ch % 5) << 6;
        // ---- cooperative fill of B chunk (64 k x 32 n) ----
        for (int e = tid; e < 2048; e += 640) {
            const int kl = e >> 5, nl = e & 31;
            const int c   = c0 + kl;
            const int pix = pix0 + nl;
            const float* xc = x + ((size_t)(b0 * CC + c) << 12);
            float val;
            if (DEFORM) {
                const int4   id = tabI[(kk << 12) + pix];
                const float4 w4 = tabW[(kk << 12) + pix];
                val = w4.x * xc[id.x] + w4.y * xc[id.y] +
                      w4.z * xc[id.z] + w4.w * xc[id.w];
            } else {
                const int y = pix >> 6, xq = pix & 63;
                const int yy = y + kk / 3 - 1;
                const int xx = xq + kk % 3 - 1;
                val = (((unsigned)yy < 64u) && ((unsigned)xx < 64u))
                          ? xc[(yy << 6) + xx] : 0.0f;
            }
            Blds[nl][kl] = val;
        }
        __syncthreads();
        // ---- 16 WMMA K-steps; one A fragment feeds two N-tiles ----
        const float* Ap = Wrt + (size_t)(kk * CC + c0) * OO + (o0 + l15);
#pragma unroll
        for (int s = 0; s < 16; ++s) {
            const int kA = (s << 2) + (half << 1);  // K = 4s + 2*half + v
            v2f a, bv0, bv1;
            a.x = Ap[(size_t)kA * OO];
            a.y = Ap[(size_t)(kA + 1) * OO];
            bv0 = *(const v2f*)(&Blds[l15][kA]);       // ds_load_b64
            bv1 = *(const v2f*)(&Blds[16 + l15][kA]);
            acc0 = __builtin_amdgcn_wmma_f32_16x16x4_f32(
                       false, a, false, bv0, (short)0, acc0, false, false);
            acc1 = __builtin_amdgcn_wmma_f32_16x16x4_f32(
                       false, a, false, bv1, (short)0, acc1, false, false);
        }
        __syncthreads();
    }
    // epilogue: D vgpr r -> row (r + 8*half); add bias, store both halves
#pragma unroll
    for (int r = 0; r < 8; ++r) {
        const int o = o0 + r + (half << 3);
        const float bo = bias[o];
        const size_t base = ((size_t)(b0 * OO + o) << 12) + pix0;
        dst[base + l15]      = acc0[r] + bo;
        dst[base + 16 + l15] = acc1[r] + bo;
    }
}

// ---------------------------------------------------------------------------
// Fused LoRA GEMM: acc = Mt' x cat (K=640; k<320 -> dconv, else h),
// then out = h + scale * acc.  B-fill is a pure copy -> async Global->LDS.
// ---------------------------------------------------------------------------
__global__ __launch_bounds__(640)
void fuse_gemm_kernel(const float* __restrict__ hbuf,
                      const float* __restrict__ dconv,
                      const float* __restrict__ Mt,
                      const float* __restrict__ scale,
                      float* __restrict__ out) {
    __shared__ float Blds[32][68];
    const int tid  = threadIdx.x;
    const int lane = tid & 31;
    const int wave = tid >> 5;
    const int half = lane >> 4;
    const int l15  = lane & 15;
    const int o0   = wave << 4;
    const int n0   = blockIdx.x << 5;
    const int b0   = n0 >> 12;
    const int pix0 = n0 & 4095;

    v8f acc0 = {}, acc1 = {};
    for (int ch = 0; ch < 10; ++ch) {
        const int c0 = ch << 6;
        for (int e = tid; e < 2048; e += 640) {
            const int kl = e >> 5, nl = e & 31;
            const int k  = c0 + kl;
            const float* src = (k < 320) ? dconv : hbuf;
            const int cc = (k < 320) ? k : (k - 320);
            const float* gp = src + (((size_t)(b0 * OO + cc) << 12) + pix0 + nl);
#if HAVE_ASYNC_LDS
            __builtin_amdgcn_global_load_async_to_lds_b32(
                (as1_int*)gp, (as3_int*)&Blds[nl][kl], 0, 0);
#else
            Blds[nl][kl] = *gp;
#endif
        }
#if HAVE_ASYNC_LDS
        __builtin_amdgcn_s_wait_asynccnt(0);
#endif
        __syncthreads();
        const float* Ap = Mt + (size_t)c0 * OO + (o0 + l15);
#pragma unroll
        for (int s = 0; s < 16; ++s) {
            const int kA = (s << 2) + (half << 1);
            v2f a, bv0, bv1;
            a.x = Ap[(size_t)kA * OO];
            a.y = Ap[(size_t)(kA + 1) * OO];
            bv0 = *(const v2f*)(&Blds[l15][kA]);
            bv1 = *(const v2f*)(&Blds[16 + l15][kA]);
            acc0 = __builtin_amdgcn_wmma_f32_16x16x4_f32(
                       false, a, false, bv0, (short)0, acc0, false, false);
            acc1 = __builtin_amdgcn_wmma_f32_16x16x4_f32(
                       false, a, false, bv1, (short)0, acc1, false, false);
        }
        __syncthreads();
    }
    const float sc = scale[0];
#pragma unroll
    for (int r = 0; r < 8; ++r) {
        const int o = o0 + r + (half << 3);
        const size_t base = ((size_t)(b0 * OO + o) << 12) + pix0;
        out[base + l15]      = hbuf[base + l15]      + sc * acc0[r];
        out[base + 16 + l15] = hbuf[base + 16 + l15] + sc * acc1[r];
    }
}

// ---------------------------------------------------------------------------
extern "C" void kernel_launch(void* const* d_in, const int* in_sizes, int n_in,
                              void* d_out, int out_size, void* d_ws, size_t ws_size,
                              hipStream_t stream) {
    const float* x      = (const float*)d_in[0];
    const float* weight = (const float*)d_in[1];
    const float* bias   = (const float*)d_in[2];
    const float* w_down = (const float*)d_in[3];
    const float* w_up   = (const float*)d_in[4];
    const float* scale  = (const float*)d_in[5];
    const float* offset = (const float*)d_in[6];

    const size_t HBYTES = (size_t)BB * OO * HWPIX * sizeof(float); // 21 MB
    char* p = (char*)d_ws;
    float* h_buf = (float*)p; p += HBYTES;
    float* dconv = (float*)p; p += HBYTES;
    float* Wrt   = (float*)p; p += (size_t)KCONV * OO * sizeof(float);
    float* Mt    = (float*)p; p += (size_t)640 * OO * sizeof(float);
    int4*  tabI  = (int4*)p;  p += (size_t)9 * HWPIX * sizeof(int4);
    float4* tabW = (float4*)p;

    build_tables_kernel<<<(9 * HWPIX + 255) / 256, 256, 0, stream>>>(offset, tabI, tabW);
    reorder_weight_kernel<<<(KCONV * OO + 255) / 256, 256, 0, stream>>>(weight, Wrt);
    build_Mt_kernel<<<(640 * OO + 255) / 256, 256, 0, stream>>>(w_down, w_up, Mt);

    conv_gemm_kernel<0><<<NTOT / 32, 640, 0, stream>>>(x, Wrt, bias, tabI, tabW, h_buf);
    conv_gemm_kernel<1><<<NTOT / 32, 640, 0, stream>>>(x, Wrt, bias, tabI, tabW, dconv);
    fuse_gemm_kernel<<<NTOT / 32, 640, 0, stream>>>(h_buf, dconv, Mt, scale, (float*)d_out);
}